// EdgeScorer_85134841741341
// MI455X (gfx1250) — compile-verified
//
#include <hip/hip_runtime.h>
#include <math.h>

#define B_   8
#define K_   20000
#define C_   256
#define E_   100000
#define HID_ 128
#define ZRAW 261      // 2*HID + 5 geom
#define ZDIM 264      // padded; col 261 = bias column, 262/263 zero
#define ZSTRIDE 268   // LDS row stride (floats): 12*r mod 64 distinct for r<16
#define ASTRIDE 260   // LDS row stride (floats): 4*r mod 64 distinct for r<16

typedef float v2f __attribute__((ext_vector_type(2)));
typedef float v4f __attribute__((ext_vector_type(4)));
typedef float v8f __attribute__((ext_vector_type(8)));

// ---------------- Kernel 0: per-graph geometry scale ----------------
__global__ void scale_kernel(const float* __restrict__ xy, float* __restrict__ scale) {
  __shared__ float sx0[256], sx1[256], sy0[256], sy1[256];
  const int b = blockIdx.x, t = threadIdx.x;
  const float* p = xy + (size_t)b * K_ * 2;
  float xmin = 3.4e38f, xmax = -3.4e38f, ymin = 3.4e38f, ymax = -3.4e38f;
  for (int k = t; k < K_; k += 256) {
    float x = p[2 * k], y = p[2 * k + 1];
    xmin = fminf(xmin, x); xmax = fmaxf(xmax, x);
    ymin = fminf(ymin, y); ymax = fmaxf(ymax, y);
  }
  sx0[t] = xmin; sx1[t] = xmax; sy0[t] = ymin; sy1[t] = ymax;
  __syncthreads();
  for (int s = 128; s > 0; s >>= 1) {
    if (t < s) {
      sx0[t] = fminf(sx0[t], sx0[t + s]); sx1[t] = fmaxf(sx1[t], sx1[t + s]);
      sy0[t] = fminf(sy0[t], sy0[t + s]); sy1[t] = fmaxf(sy1[t], sy1[t + s]);
    }
    __syncthreads();
  }
  if (t == 0) {
    float sc = fmaxf(sx1[0] - sx0[0] + 1.0f, sy1[0] - sy0[0] + 1.0f);
    scale[b] = fmaxf(sc / 8.0f, 1.0f);
  }
}

// ---------------- Kernel 1: pad W1 + fold b1 into bias column ----------------
__global__ void w1pad_kernel(const float* __restrict__ W1, const float* __restrict__ b1,
                             float* __restrict__ W1p) {
  int i = blockIdx.x * 256 + threadIdx.x;           // over 128*264
  if (i >= HID_ * ZDIM) return;
  int n = i / ZDIM, k = i % ZDIM;
  float v = 0.0f;
  if (k < ZRAW)       v = W1[n * ZRAW + k];
  else if (k == ZRAW) v = b1[n];                    // bias column
  W1p[i] = v;
}

// ---------------- Kernel 2: h = node_feat @ W_lin.T (fp32 WMMA) ----------------
// grid = (K/16, B); 256 threads = 8 waves; wave w owns N-tile w (HID = 8*16).
__global__ void proj_kernel(const float* __restrict__ nf, const float* __restrict__ Wl,
                            float* __restrict__ h) {
  __shared__ __align__(16) float As[16][ASTRIDE];
  const int tm = blockIdx.x, b = blockIdx.y;
  const int t = threadIdx.x, wave = t >> 5, lane = t & 31;
  const float* src = nf + ((size_t)b * K_ + (size_t)tm * 16) * C_;

  // stage 16x256 A-tile into LDS (1024 float4 loads, 4 per thread)
  for (int j = 0; j < 4; ++j) {
    int chunk = t + j * 256;          // 0..1023
    int row = chunk >> 6;             // chunk / 64
    int c4  = chunk & 63;
    v4f v = *reinterpret_cast<const v4f*>(src + row * C_ + c4 * 4);
    *reinterpret_cast<v4f*>(&As[row][c4 * 4]) = v;
  }
  __syncthreads();

  const int r    = lane & 15;
  const int koff = (lane >> 4) * 2;   // lanes 0-15: K=k0,k0+1 ; lanes 16-31: K=k0+2,k0+3
  const int n    = wave * 16 + r;
  v8f acc = {};
  for (int kk = 0; kk < C_; kk += 4) {
    v2f a; a.x = As[r][kk + koff]; a.y = As[r][kk + koff + 1];
    v2f bf = *reinterpret_cast<const v2f*>(Wl + n * C_ + kk + koff);
    acc = __builtin_amdgcn_wmma_f32_16x16x4_f32(false, a, false, bf, (short)0, acc,
                                                false, false);
  }
  // D layout: vgpr j -> M = j + 8*(lane>=16), N = lane&15
  float* hb = h + ((size_t)b * K_ + (size_t)tm * 16) * HID_ + wave * 16;
  const int half = lane >> 4;
  for (int j = 0; j < 8; ++j)
    hb[(j + half * 8) * HID_ + r] = acc[j];
}

// ---------------- Kernel 3: edge MLP (gather + geom + fp32 WMMA + reduce) ----------------
// grid = (E/16, B); 256 threads = 8 waves; each block handles 16 edges.
__global__ void edge_kernel(const int* __restrict__ edges, const float* __restrict__ xy,
                            const float* __restrict__ h, const float* __restrict__ W1p,
                            const float* __restrict__ W2, const float* __restrict__ b2,
                            const float* __restrict__ scale, float* __restrict__ out) {
  __shared__ __align__(16) float Z[16][ZSTRIDE];
  __shared__ float accum[16];
  const int tile = blockIdx.x, b = blockIdx.y;
  const int t = threadIdx.x, wave = t >> 5, lane = t & 31;
  const int* eb = edges + (size_t)b * 2 * E_;
  const int e0 = tile * 16;

  // geom features + bias column + accumulator init (threads 0..15)
  if (t < 16) {
    int e = e0 + t;
    int s = eb[e], d = eb[E_ + e];
    const float* xb = xy + (size_t)b * K_ * 2;
    float xs = xb[2 * s], ys = xb[2 * s + 1];
    float xd = xb[2 * d], yd = xb[2 * d + 1];
    float dx = xd - xs, dy = yd - ys;
    float dist = fminf(sqrtf(dx * dx + dy * dy) + 1e-6f, 1e6f);
    float ang  = atan2f(dy, dx + 1e-6f);
    float sc   = scale[b];
    Z[t][256] = dx / sc;
    Z[t][257] = dy / sc;
    Z[t][258] = dist / sc;
    Z[t][259] = cosf(ang);
    Z[t][260] = sinf(ang);
    Z[t][261] = 1.0f;      // bias column (matches b1 folded into W1p)
    Z[t][262] = 0.0f;
    Z[t][263] = 0.0f;
    accum[t] = 0.0f;
  }

  // gather hs / hd: thread t handles edge i = t/16, 8 floats per half
  {
    int i = t >> 4, part = t & 15;
    int e = e0 + i;
    int s = eb[e], d = eb[E_ + e];
    const float* hs = h + ((size_t)b * K_ + (size_t)s) * HID_ + part * 8;
    const float* hd = h + ((size_t)b * K_ + (size_t)d) * HID_ + part * 8;
    v4f a0 = *reinterpret_cast<const v4f*>(hs);
    v4f a1 = *reinterpret_cast<const v4f*>(hs + 4);
    v4f c0 = *reinterpret_cast<const v4f*>(hd);
    v4f c1 = *reinterpret_cast<const v4f*>(hd + 4);
    *reinterpret_cast<v4f*>(&Z[i][part * 8])       = a0;
    *reinterpret_cast<v4f*>(&Z[i][part * 8 + 4])   = a1;
    *reinterpret_cast<v4f*>(&Z[i][128 + part * 8])     = c0;
    *reinterpret_cast<v4f*>(&Z[i][128 + part * 8 + 4]) = c1;
  }
  __syncthreads();

  // hmid(16x128) = relu(z @ W1p.T); wave w owns N-tile w
  const int r    = lane & 15;
  const int koff = (lane >> 4) * 2;
  const int n    = wave * 16 + r;
  v8f acc = {};
  for (int kk = 0; kk < ZDIM; kk += 4) {
    v2f a; a.x = Z[r][kk + koff]; a.y = Z[r][kk + koff + 1];
    v2f bf = *reinterpret_cast<const v2f*>(W1p + n * ZDIM + kk + koff);
    acc = __builtin_amdgcn_wmma_f32_16x16x4_f32(false, a, false, bf, (short)0, acc,
                                                false, false);
  }

  // out[e] = sum_n relu(hmid[e][n]) * W2[n]; reduce 16 lanes per half-wave
  const float w2v = W2[n];
  const int half = lane >> 4;
  for (int j = 0; j < 8; ++j) {
    float v = fmaxf(acc[j], 0.0f) * w2v;
    v += __shfl_xor(v, 1, 32);
    v += __shfl_xor(v, 2, 32);
    v += __shfl_xor(v, 4, 32);
    v += __shfl_xor(v, 8, 32);
    if (r == 0) atomicAdd(&accum[j + half * 8], v);   // ds_add_f32
  }
  __syncthreads();
  if (t < 16)
    out[(size_t)b * E_ + e0 + t] = accum[t] + b2[0];
}

// ---------------- launch ----------------
extern "C" void kernel_launch(void* const* d_in, const int* in_sizes, int n_in,
                              void* d_out, int out_size, void* d_ws, size_t ws_size,
                              hipStream_t stream) {
  const float* node_feat = (const float*)d_in[0];
  const int*   edges     = (const int*)  d_in[1];
  const float* node_xy   = (const float*)d_in[2];
  const float* W_lin     = (const float*)d_in[3];
  const float* W1        = (const float*)d_in[4];
  const float* b1        = (const float*)d_in[5];
  const float* W2        = (const float*)d_in[6];
  const float* b2        = (const float*)d_in[7];
  float* out = (float*)d_out;

  char* ws = (char*)d_ws;
  float* scale = (float*)ws;                                   // 8 floats
  float* W1p   = (float*)(ws + 256);                           // 128*264 floats
  float* h     = (float*)(ws + 256 + (size_t)HID_ * ZDIM * sizeof(float)); // B*K*HID floats

  scale_kernel<<<dim3(B_), 256, 0, stream>>>(node_xy, scale);
  w1pad_kernel<<<dim3((HID_ * ZDIM + 255) / 256), 256, 0, stream>>>(W1, b1, W1p);
  proj_kernel<<<dim3(K_ / 16, B_), 256, 0, stream>>>(node_feat, W_lin, h);
  edge_kernel<<<dim3(E_ / 16, B_), 256, 0, stream>>>(edges, node_xy, h, W1p, W2, b2,
                                                     scale, out);
}